// Decoder_74835510165950
// MI455X (gfx1250) — compile-verified
//
#include <hip/hip_runtime.h>
#include <hip/hip_bf16.h>
#include <math.h>

// ---------------------------------------------------------------------------
// Fused pointer-network decoder for MI455X (gfx1250, wave32).
//
// Refactoring: q_len==1 attention means
//   scores[h,n]   = U[h]  . x[n]      U = (q-slices @ Wk-slices)/4   (8x128/batch)
//   att_out[h]    = Wv_h @ M[h]/s_h   M[h] = sum_n a[h,n] x[n]       (8x128/batch)
//   ptr_score[n]  = t . x[n]          t = (final_q @ logit_Wk)/sqrt(128)
// so patch_embeddings (65MB) is streamed exactly twice (2nd pass hits L2),
// and Wk/Wv/logit_Wk projections are never materialized.
//
// Pass A uses V_WMMA_F32_16X16X4_F32 (fp32-exact):
//   S(16x16)  = U(16x128) x Xtile^T      -- 32 wmma / 16-row tile
//   M(16x128) += A(16x16 weights) x Xtile -- 32 wmma / 16-row tile
// with flash-style online softmax per wave; partials merged in k_combine.
//
// Workspace (floats): U[B*2048] | P[B*32*1040] | T[B*128] | psum[B]  (~9.1MB)
// ---------------------------------------------------------------------------

#define HDIM 128
#define NP   2000
#define WPB  4        // waves per block in pass A
#define NBLK 8        // blocks per batch in pass A  (NBLK*WPB = 32 wave-partials)
#define PB   4        // blocks per batch in pointer pass

typedef float v2f __attribute__((ext_vector_type(2)));
typedef float v8f __attribute__((ext_vector_type(8)));

// K1: per-batch glimpse_q and U = (q . Wk-head-slices)/sqrt(16); zero psum.
__global__ __launch_bounds__(128) void k_setup(
    const float* __restrict__ x, const float* __restrict__ cls,
    const float* __restrict__ Wq_g, const float* __restrict__ Wq_f,
    const float* __restrict__ Wq_l, const float* __restrict__ Wk,
    const int* __restrict__ last_patch,
    float* __restrict__ wsU, float* __restrict__ psum)
{
  int b = blockIdx.x, j = threadIdx.x;
  __shared__ float sc[HDIM], sl[HDIM], sq[HDIM];
  int lp = last_patch[b];
  sc[j] = cls[b * HDIM + j];
  sl[j] = x[((size_t)b * NP + lp) * HDIM + j];
  __syncthreads();
  float acc = 0.f;
  const float* wg = Wq_g + j * HDIM;
  const float* wf = Wq_f + j * HDIM;
  const float* wl = Wq_l + j * HDIM;
  for (int i = 0; i < HDIM; ++i) acc += sc[i] * wg[i] + sl[i] * (wf[i] + wl[i]);
  sq[j] = acc;
  __syncthreads();
  float* U = wsU + (size_t)b * 2048;
  for (int h = 0; h < 8; ++h) {
    float u = 0.f;
    for (int d = 0; d < 16; ++d) u += sq[h * 16 + d] * Wk[(h * 16 + d) * HDIM + j];
    U[h * HDIM + j]       = 0.25f * u;  // fold 1/sqrt(head_dim)
    U[(h + 8) * HDIM + j] = 0.f;        // pad heads 8..15 with zeros
  }
  if (j == 0) psum[b] = 0.f;
}

// K2: streaming attention pass. One 16-row x tile per wave iteration.
__global__ __launch_bounds__(128) void k_attn(
    const float* __restrict__ x, const float* __restrict__ wsU,
    const int* __restrict__ last_patch, float* __restrict__ wsP)
{
  int blk  = blockIdx.x;
  int b    = blk / NBLK, bb = blk % NBLK;
  int wave = threadIdx.x >> 5, lane = threadIdx.x & 31;
  int l16  = lane & 15, hi = lane >> 4;
  int lp   = last_patch[b];

  // per-wave private LDS regions (padded strides: 132 -> conflict-free + 16B
  // aligned float4 rows; 20 -> conflict-free score transpose)
  __shared__ float xs[WPB][16 * 132];
  __shared__ float sw[WPB][16 * 20];
  __shared__ float scl[WPB][16];
  float* myxs  = xs[wave];
  float* mysw  = sw[wave];
  float* myscl = scl[wave];

  // Preload U in WMMA A-operand layout: A(16x4) element (M,K):
  //   M = lane%16, K = vgpr + 2*(lane>=16); K-step k covers K=4k..4k+3.
  const float* U = wsU + (size_t)b * 2048;
  v2f ua[32];
  for (int k = 0; k < 32; ++k) {
    const float* up = U + l16 * HDIM + 4 * k + 2 * hi;
    ua[k].x = up[0];
    ua[k].y = up[1];
  }

  v8f zero = {};
  v8f mc[8];                       // M accumulator, C layout: row r+8*hi, col c*16+l16
  for (int c = 0; c < 8; ++c) mc[c] = zero;
  float m_run = -3.0e38f, s_run = 0.f;

  int gw = bb * WPB + wave;        // 0..31 global wave id within batch
  for (int tile = gw; tile < NP / 16; tile += NBLK * WPB) {
    int n0 = tile * 16;
    // 1) stage 16x128 x tile into LDS (coalesced b128 loads)
    const float* xt = x + ((size_t)b * NP + n0) * HDIM;
    for (int r = 0; r < 16; ++r) {
      float4 v = *(const float4*)(xt + r * HDIM + lane * 4);
      *(float4*)(myxs + r * 132 + lane * 4) = v;
    }
    __builtin_amdgcn_wave_barrier();

    // 2) S(16 heads x 16 n) = U x Xtile^T  (K=128 -> 32 wmma)
    v8f c = zero;
    for (int k = 0; k < 32; ++k) {
      int i = 4 * k + 2 * hi;
      v2f bv;
      bv.x = myxs[l16 * 132 + i];      // B(K=i,   N=l16) = x[n=l16][i]
      bv.y = myxs[l16 * 132 + i + 1];  // B(K=i+1, N=l16)
      c = __builtin_amdgcn_wmma_f32_16x16x4_f32(false, ua[k], false, bv,
                                                (short)0, c, false, false);
    }
    // 3) transpose scores to sw[n][h] (C layout: c[r] = S[r+8*hi][l16])
    for (int r = 0; r < 8; ++r) mysw[l16 * 20 + 8 * hi + r] = c[r];
    __builtin_amdgcn_wave_barrier();

    // 4) online softmax; lane handles head h=l16 (both halves duplicate)
    float tmax = -3.0e38f;
    for (int n = 0; n < 16; ++n) tmax = fmaxf(tmax, mysw[n * 20 + l16]);
    float nm      = fmaxf(m_run, tmax);
    float rescale = __expf(m_run - nm);
    m_run = nm;
    s_run *= rescale;
    for (int n = 0; n < 16; ++n) {
      float e = __expf(mysw[n * 20 + l16] - nm);
      if (n0 + n == lp) e = 0.f;     // mask just-visited patch (== -1e9 mask)
      mysw[n * 20 + l16] = e;        // weights, [n][h]
      s_run += e;
    }
    if (hi == 0) myscl[l16] = rescale;
    __builtin_amdgcn_wave_barrier();

    // rescale M accumulators per their C-layout head row
    float scr[8];
    for (int r = 0; r < 8; ++r) scr[r] = myscl[r + 8 * hi];
    for (int ch = 0; ch < 8; ++ch)
      for (int r = 0; r < 8; ++r) mc[ch][r] *= scr[r];
    __builtin_amdgcn_wave_barrier();

    // 5) M(16x128) += W(16 heads x 16 n) x Xtile   (K=16 -> 4 steps x 8 chunks)
    for (int kk = 0; kk < 4; ++kk) {
      int na = 4 * kk + 2 * hi;      // K index (patch-in-tile)
      v2f a;
      a.x = mysw[na * 20 + l16];       // A(M=l16, K=na)
      a.y = mysw[(na + 1) * 20 + l16]; // A(M=l16, K=na+1)
      for (int ch = 0; ch < 8; ++ch) {
        v2f bv;
        bv.x = myxs[na * 132 + ch * 16 + l16];        // B(K=na,   N=col)
        bv.y = myxs[(na + 1) * 132 + ch * 16 + l16];  // B(K=na+1, N=col)
        mc[ch] = __builtin_amdgcn_wmma_f32_16x16x4_f32(false, a, false, bv,
                                                       (short)0, mc[ch],
                                                       false, false);
      }
    }
    __builtin_amdgcn_wave_barrier();
  }

  // write wave partial: m[8], s[8], M[8][128] (real heads live in hi==0 rows)
  float* P = wsP + ((size_t)b * 32 + gw) * 1040;
  if (hi == 0 && l16 < 8) { P[l16] = m_run; P[8 + l16] = s_run; }
  if (hi == 0)
    for (int ch = 0; ch < 8; ++ch)
      for (int r = 0; r < 8; ++r)
        P[16 + r * 128 + ch * 16 + l16] = mc[ch][r];
}

// K3: merge 32 wave-partials per batch, finish att_out -> final_q -> t.
__global__ __launch_bounds__(128) void k_combine(
    const float* __restrict__ wsP, const float* __restrict__ Wv,
    const float* __restrict__ Wc, const float* __restrict__ bcv,
    const float* __restrict__ lWk, float* __restrict__ wsT)
{
  int b = blockIdx.x, j = threadIdx.x;
  __shared__ float efs[32][8];
  __shared__ float sinv[8];
  __shared__ float Mn[8][HDIM];
  __shared__ float o[HDIM];
  __shared__ float fq[HDIM];
  const float* P = wsP + (size_t)b * 32 * 1040;
  if (j < 8) {
    float gm = -3.0e38f;
    for (int p = 0; p < 32; ++p) gm = fmaxf(gm, P[p * 1040 + j]);
    float st = 0.f;
    for (int p = 0; p < 32; ++p) {
      float e = __expf(P[p * 1040 + j] - gm);
      efs[p][j] = e;
      st += e * P[p * 1040 + 8 + j];
    }
    sinv[j] = 1.f / st;
  }
  __syncthreads();
  for (int h = 0; h < 8; ++h) {     // thread j = column i
    float acc = 0.f;
    for (int p = 0; p < 32; ++p) acc += P[p * 1040 + 16 + h * 128 + j] * efs[p][h];
    Mn[h][j] = acc * sinv[h];
  }
  __syncthreads();
  {                                  // att_out[j] = Wv[j] . Mn[j/16]
    int h = j >> 4;
    float acc = 0.f;
    const float* wv = Wv + j * HDIM;
    for (int i = 0; i < HDIM; ++i) acc += wv[i] * Mn[h][i];
    o[j] = acc;
  }
  __syncthreads();
  {                                  // final_q = o @ Wc.T + bc
    float acc = bcv[j];
    const float* wc = Wc + j * HDIM;
    for (int i = 0; i < HDIM; ++i) acc += wc[i] * o[i];
    fq[j] = acc;
  }
  __syncthreads();
  {                                  // t = (fq @ logit_Wk)/sqrt(128)
    float acc = 0.f;
    for (int j2 = 0; j2 < HDIM; ++j2) acc += fq[j2] * lWk[j2 * HDIM + j];
    wsT[b * HDIM + j] = acc * 0.08838834764831845f;
  }
}

// K4: pointer pass. score ptrs bounded by |10*tanh|<=10 -> fixed shift, no max.
__global__ __launch_bounds__(256) void k_pointer(
    const float* __restrict__ x, const float* __restrict__ wsT,
    const int* __restrict__ last_patch, float* __restrict__ out,
    float* __restrict__ psum)
{
  int blk  = blockIdx.x;
  int b    = blk / PB, pb = blk % PB;
  int wave = threadIdx.x >> 5, lane = threadIdx.x & 31;
  __shared__ float ts[HDIM];
  if (threadIdx.x < HDIM) ts[threadIdx.x] = wsT[b * HDIM + threadIdx.x];
  __syncthreads();
  int lp = last_patch[b];
  int gw = pb * 8 + wave;
  float wsum = 0.f;
  for (int n = gw; n < NP; n += PB * 8) {
    const float* xr = x + ((size_t)b * NP + n) * HDIM;
    float p = 0.f;
    for (int q = 0; q < 4; ++q) p += xr[lane + 32 * q] * ts[lane + 32 * q];
    for (int off = 16; off > 0; off >>= 1) p += __shfl_xor(p, off, 32);
    if (lane == 0) {
      float e = (n == lp) ? 0.f : __expf(10.f * tanhf(p) - 10.f);
      out[(size_t)b * NP + n] = e;
      wsum += e;
    }
  }
  if (lane == 0) atomicAdd(&psum[b], wsum);
}

// K5: normalize softmax.
__global__ __launch_bounds__(256) void k_norm(float* __restrict__ out,
                                              const float* __restrict__ psum)
{
  int b = blockIdx.x;
  float inv = 1.f / psum[b];
  for (int n = threadIdx.x; n < NP; n += 256) out[(size_t)b * NP + n] *= inv;
}

extern "C" void kernel_launch(void* const* d_in, const int* in_sizes, int n_in,
                              void* d_out, int out_size, void* d_ws, size_t ws_size,
                              hipStream_t stream) {
  const float* x    = (const float*)d_in[0];
  const float* cls  = (const float*)d_in[1];
  const float* Wq_g = (const float*)d_in[2];
  const float* Wq_f = (const float*)d_in[3];
  const float* Wq_l = (const float*)d_in[4];
  const float* Wk   = (const float*)d_in[5];
  const float* Wv   = (const float*)d_in[6];
  const float* lWk  = (const float*)d_in[7];
  const float* Wc   = (const float*)d_in[8];
  const float* bc   = (const float*)d_in[9];
  const int*   lp   = (const int*)d_in[10];
  int B = in_sizes[1] / HDIM;                  // fixed_content_cls: (B,1,H)

  float* ws   = (float*)d_ws;
  float* wsU  = ws;                            // B*2048
  float* wsP  = wsU + (size_t)B * 2048;        // B*32*1040
  float* wsT  = wsP + (size_t)B * 32 * 1040;   // B*128
  float* psum = wsT + (size_t)B * HDIM;        // B
  float* out  = (float*)d_out;

  k_setup<<<B, 128, 0, stream>>>(x, cls, Wq_g, Wq_f, Wq_l, Wk, lp, wsU, psum);
  k_attn<<<B * NBLK, 32 * WPB, 0, stream>>>(x, wsU, lp, wsP);
  k_combine<<<B, 128, 0, stream>>>(wsP, Wv, Wc, bc, lWk, wsT);
  k_pointer<<<B * PB, 256, 0, stream>>>(x, wsT, lp, out, psum);
  k_norm<<<B, 256, 0, stream>>>(out, psum);
}